// Linear_83648783057672
// MI455X (gfx1250) — compile-verified
//
#include <hip/hip_runtime.h>
#include <cmath>

typedef __attribute__((ext_vector_type(8))) float v8f;
typedef __attribute__((ext_vector_type(2))) float v2f;

constexpr int IN_DIM = 592;
constexpr int BATCH  = 100000;

// Grouped irrep linear: each wave owns one 16-row M tile and all N tiles of its
// block, accumulating with V_WMMA_F32_16X16X4_F32 (K stepped by 4, fully
// unrolled). W block staged in LDS in K-pair-interleaved layout so every
// B fragment is a single aligned ds_load_b64. x/out are the only HBM traffic
// (~474 MB total -> ~20 us roofline at 23.3 TB/s), streamed non-temporally.
template <int MUL, int DIM>
__global__ __launch_bounds__(256) void irrep_linear_kernel(
    const float* __restrict__ x,
    const float* __restrict__ w,
    float* __restrict__ out,
    int xcol,          // column offset of this block in the 592-wide row
    int wofs,          // flat offset of this block's MULxMUL weight matrix
    float pw)          // mul^-0.5 path weight
{
    // K-pair-interleaved weights: sW[(k>>1)*(2*MUL) + 2*n + (k&1)] = pw*W[k,n]
    __shared__ float sW[MUL * MUL];

    const int tid = threadIdx.x;
    for (int idx = tid; idx < MUL * MUL; idx += 256) {
        const int k = idx / MUL;              // MUL is a power of two -> shifts
        const int n = idx % MUL;
        sW[(k >> 1) * (2 * MUL) + 2 * n + (k & 1)] = w[wofs + idx] * pw;
    }
    __syncthreads();

    const int lane    = tid & 31;
    const int wave    = tid >> 5;
    const int lan16   = lane & 15;
    const int halfSel = lane >> 4;            // 0: lanes 0-15, 1: lanes 16-31

    const int rows      = BATCH * DIM;        // fused (z,i) row count
    const int numMTiles = rows >> 4;          // exact: rows % 16 == 0
    const int mTile     = blockIdx.x * 8 + wave;
    if (mTile >= numMTiles) return;           // whole-wave guard: EXEC all-1s

    const int mBase = mTile * 16;
    constexpr int NT = MUL / 16;              // N tiles per wave

    // Per-lane loop-invariant base pointers; all loop offsets become immediates.
    const int mRow = mBase + lan16;
    const int zA   = mRow / DIM;
    const int iA   = mRow % DIM;
    const float* ap = x + zA * IN_DIM + xcol + iA + 2 * halfSel * DIM;
    const float* bp = &sW[halfSel * (2 * MUL) + 2 * lan16];

    v8f acc[NT] = {};

#pragma unroll
    for (int k0 = 0; k0 < MUL; k0 += 4) {
        v2f a;                                // A frag: VGPR0=K(k0+2h), VGPR1=K+1
        a.x = __builtin_nontemporal_load(ap + k0 * DIM);
        a.y = __builtin_nontemporal_load(ap + k0 * DIM + DIM);
#pragma unroll
        for (int t = 0; t < NT; ++t) {
            const v2f b = *(const v2f*)(bp + k0 * MUL + t * 32);  // ds_load_b64
            acc[t] = __builtin_amdgcn_wmma_f32_16x16x4_f32(
                /*neg_a=*/false, a, /*neg_b=*/false, b,
                /*c_mod=*/(short)0, acc[t],
                /*reuse_a=*/false, /*reuse_b=*/false);
        }
    }

    // C/D layout: VGPR v holds M = v (lanes 0-15) / v+8 (lanes 16-31), N = lane&15.
#pragma unroll
    for (int v = 0; v < 8; ++v) {
        const int row   = mBase + v + 8 * halfSel;
        const int z     = row / DIM;
        const int i     = row % DIM;
        float* op = out + z * IN_DIM + xcol + i + lan16 * DIM;
#pragma unroll
        for (int t = 0; t < NT; ++t) {
            __builtin_nontemporal_store(acc[t][v], op + t * 16 * DIM);
        }
    }
}

extern "C" void kernel_launch(void* const* d_in, const int* in_sizes, int n_in,
                              void* d_out, int out_size, void* d_ws, size_t ws_size,
                              hipStream_t stream) {
    (void)in_sizes; (void)n_in; (void)d_ws; (void)ws_size; (void)out_size;
    const float* x = (const float*)d_in[0];
    const float* w = (const float*)d_in[1];
    float* out = (float*)d_out;

    // Block 0: mul=128, dim=1  (xcol 0,   wofs 0)
    {
        const int tiles = (BATCH * 1) / 16;            // 6250
        const int wgs   = (tiles + 7) / 8;
        irrep_linear_kernel<128, 1><<<wgs, 256, 0, stream>>>(
            x, w, out, 0, 0, 1.0f / sqrtf(128.0f));
    }
    // Block 1: mul=64, dim=3   (xcol 128, wofs 16384)
    {
        const int tiles = (BATCH * 3) / 16;            // 18750
        const int wgs   = (tiles + 7) / 8;
        irrep_linear_kernel<64, 3><<<wgs, 256, 0, stream>>>(
            x, w, out, 128, 16384, 1.0f / sqrtf(64.0f));
    }
    // Block 2: mul=32, dim=5   (xcol 320, wofs 20480)
    {
        const int tiles = (BATCH * 5) / 16;            // 31250
        const int wgs   = (tiles + 7) / 8;
        irrep_linear_kernel<32, 5><<<wgs, 256, 0, stream>>>(
            x, w, out, 320, 20480, 1.0f / sqrtf(32.0f));
    }
    // Block 3: mul=16, dim=7   (xcol 480, wofs 21504)
    {
        const int tiles = (BATCH * 7) / 16;            // 43750
        const int wgs   = (tiles + 7) / 8;
        irrep_linear_kernel<16, 7><<<wgs, 256, 0, stream>>>(
            x, w, out, 480, 21504, 1.0f / sqrtf(16.0f));
    }
}